// BinancePerpStructuralLoss_2138893713741
// MI455X (gfx1250) — compile-verified
//
#include <hip/hip_runtime.h>
#include <stdint.h>

// Problem constants (from reference): F = 2*12*10 = 240 features/row,
// C = 228 constraints, B = 128 (mean divisor). Rows = B*T derived at launch.
#define F_DIM   240
#define MAXJ    8          // ceil(228/32) constraint pairs per lane
#define BLOCK   256        // 8 waves (wave32)
#define GRID    1024
#define WPB     (BLOCK / 32)
#define SLAB_F  256        // padded floats per LDS slab (row is 240)
#define INV_B   (1.0f / 128.0f)

// ---------------------------------------------------------------------------
// Main kernel: one wave per row, async-copy row -> LDS (double buffered),
// gather + relu + accumulate, hierarchical reduce to per-block partial.
// ---------------------------------------------------------------------------
__global__ __launch_bounds__(BLOCK) void structural_loss_main(
    const float* __restrict__ pred,
    const int*   __restrict__ idx_a,
    const int*   __restrict__ idx_b,
    int n_constraints, int nrows,
    float* __restrict__ block_partials)
{
    __shared__ __align__(16) float smem[WPB * 2 * SLAB_F];  // 16 KB
    __shared__ float wsum[WPB];

    const int lane = threadIdx.x & 31;
    const int wid  = threadIdx.x >> 5;

    // Per-lane constraint indices in registers; pad with (0,0): relu(0)=0.
    int ca[MAXJ], cb[MAXJ];
#pragma unroll
    for (int j = 0; j < MAXJ; ++j) {
        int c = (j << 5) + lane;
        if (c < n_constraints) { ca[j] = idx_a[c]; cb[j] = idx_b[c]; }
        else                   { ca[j] = 0;        cb[j] = 0;        }
    }

    // Low 32 bits of the generic pointer to LDS == byte offset in LDS space.
    const uint32_t smem_lo = (uint32_t)(uintptr_t)(&smem[0]);
    const uint32_t lane16  = (uint32_t)(lane << 4);

    const int wgid   = blockIdx.x * WPB + wid;
    const int stride = GRID * WPB;

    float acc = 0.0f;

    int r = wgid;
    int parity = 0;

    // Prologue: async-stage the first row into slab 0 of this wave.
    if (r < nrows) {
        uint64_t ga = (uint64_t)(uintptr_t)(pred + (size_t)r * F_DIM) + lane16;
        uint32_t la = smem_lo + (uint32_t)((wid << 9) << 2) + lane16;   // slab0
        asm volatile("global_load_async_to_lds_b128 %0, %1, off"
                     :: "v"(la), "v"(ga) : "memory");
        if (lane < 28)  // remaining 960-512 = 448 bytes
            asm volatile("global_load_async_to_lds_b128 %0, %1, off"
                         :: "v"(la + 512u), "v"(ga + 512ull) : "memory");
    }

    while (r < nrows) {
        int rn = r + stride;
        if (rn < nrows) {
            // Prefetch next row into the other slab, then wait for the
            // current row (ASYNCcnt completes in-order; <=2 retires row r).
            int np = parity ^ 1;
            uint64_t ga = (uint64_t)(uintptr_t)(pred + (size_t)rn * F_DIM) + lane16;
            uint32_t la = smem_lo + (uint32_t)(((wid << 9) + (np << 8)) << 2) + lane16;
            asm volatile("global_load_async_to_lds_b128 %0, %1, off"
                         :: "v"(la), "v"(ga) : "memory");
            if (lane < 28)
                asm volatile("global_load_async_to_lds_b128 %0, %1, off"
                             :: "v"(la + 512u), "v"(ga + 512ull) : "memory");
            asm volatile("s_wait_asynccnt 0x2" ::: "memory");
        } else {
            asm volatile("s_wait_asynccnt 0x0" ::: "memory");
        }

        // Gather constraints from this wave's LDS slab and accumulate relu.
        const float* slab = &smem[(wid << 9) + (parity << 8)];
#pragma unroll
        for (int j = 0; j < MAXJ; ++j) {
            float d = slab[ca[j]] - slab[cb[j]];
            acc += fmaxf(d, 0.0f);
        }

        r = rn;
        parity ^= 1;
    }

    // wave32 butterfly reduce
#pragma unroll
    for (int off = 16; off > 0; off >>= 1)
        acc += __shfl_xor(acc, off, 32);
    if (lane == 0) wsum[wid] = acc;
    __syncthreads();
    if (threadIdx.x == 0) {
        float s = 0.0f;
#pragma unroll
        for (int i = 0; i < WPB; ++i) s += wsum[i];
        block_partials[blockIdx.x] = s;
    }
}

// ---------------------------------------------------------------------------
// Deterministic final reduction of GRID partials; applies 1/B scaling.
// ---------------------------------------------------------------------------
__global__ __launch_bounds__(256) void structural_loss_reduce(
    const float* __restrict__ part, float* __restrict__ out, int n, float scale)
{
    __shared__ float sh[256];
    float s = 0.0f;
    for (int i = threadIdx.x; i < n; i += 256) s += part[i];
    sh[threadIdx.x] = s;
    __syncthreads();
    for (int st = 128; st > 0; st >>= 1) {
        if (threadIdx.x < st) sh[threadIdx.x] += sh[threadIdx.x + st];
        __syncthreads();
    }
    if (threadIdx.x == 0) out[0] = sh[0] * scale;
}

// ---------------------------------------------------------------------------
extern "C" void kernel_launch(void* const* d_in, const int* in_sizes, int n_in,
                              void* d_out, int out_size, void* d_ws, size_t ws_size,
                              hipStream_t stream)
{
    const float* pred  = (const float*)d_in[0];
    const int*   idx_a = (const int*)d_in[1];
    const int*   idx_b = (const int*)d_in[2];

    const int C     = in_sizes[1];            // 228
    const int nrows = in_sizes[0] / F_DIM;    // B*T = 131072

    float* partials = (float*)d_ws;           // GRID floats of scratch

    structural_loss_main<<<GRID, BLOCK, 0, stream>>>(pred, idx_a, idx_b, C, nrows, partials);
    structural_loss_reduce<<<1, 256, 0, stream>>>(partials, (float*)d_out, GRID, INV_B);
}